// HebbianNet_22857815949528
// MI455X (gfx1250) — compile-verified
//
#include <hip/hip_runtime.h>

typedef __attribute__((ext_vector_type(2))) float v2f;
typedef __attribute__((ext_vector_type(4))) float v4f;
typedef __attribute__((ext_vector_type(8))) float v8f;

#define RATE_CONST 0.001f
#define KMAX 1024
#define LDSPAD 8                    // row pad (floats) to decorrelate LDS banks
#define LDSK (KMAX + LDSPAD)

// One block = 128 threads = 4 waves. Each block computes a 16-wide output tile
// of out[8, O] for one Hebbian layer; the 4 waves split the K dimension and
// accumulate with V_WMMA_F32_16X16X4_F32 (f32 in / f32 out, full precision).
__global__ __launch_bounds__(128) void hebb_layer_kernel(
    const float* __restrict__ vin,   // (8, K) row-major
    const float* __restrict__ W,     // (O, K) row-major
    const float* __restrict__ bias,  // (O)
    const float* __restrict__ cw1,   // (8, 3)
    const float* __restrict__ cb1,   // (8)
    const float* __restrict__ cw2,   // (1, 8)
    const float* __restrict__ cb2,   // (1)
    const int*   __restrict__ batch_num,
    float*       __restrict__ out,   // (8, O)
    int K, int O)
{
    __shared__ float lds_v[8 * LDSK];     // padded rows: bank-conflict-free B frags
    __shared__ v8f   lds_part[4][32];     // per-wave partial 16x16 f32 tiles
    __shared__ float lds_S[8];
    __shared__ float lds_Q[8];

    const int tid   = threadIdx.x;
    const int lane  = tid & 31;
    // wave index is wave-uniform: force it into an SGPR so the K-loop bounds
    // are scalar (plain s_cbranch loop, no per-iteration exec masking).
    const int wave  = __builtin_amdgcn_readfirstlane(tid >> 5);
    const int oBase = blockIdx.x * 16;

    // ---- stage 1: cooperative load of v into LDS (vectorized b128) ----
    for (int m = 0; m < 8; ++m) {
        for (int k = tid * 4; k < K; k += 128 * 4) {
            *(v4f*)&lds_v[m * LDSK + k] = *(const v4f*)&vin[m * K + k];
        }
    }
    __syncthreads();

    // ---- stage 2: per-batch-row S = sum v, Q = sum v^2 (16 lanes per row) ----
    {
        const int row = tid >> 4;     // 0..7
        const int sub = tid & 15;
        float s = 0.f, q = 0.f;
        for (int k = sub; k < K; k += 16) {
            const float x = lds_v[row * LDSK + k];
            s += x;
            q += x * x;
        }
        for (int off = 8; off > 0; off >>= 1) {
            s += __shfl_xor(s, off, 16);
            q += __shfl_xor(q, off, 16);
        }
        if (sub == 0) { lds_S[row] = s; lds_Q[row] = q; }
    }

    // ---- collapsed 1x1-conv coefficients (uniform; scalarizes) ----
    float a0 = 0.f, a1 = 0.f, a2 = 0.f, cc = 0.f;
#pragma unroll
    for (int h = 0; h < 8; ++h) {
        const float w2 = cw2[h];
        a0 += w2 * cw1[h * 3 + 0];   // coeff of vi
        a1 += w2 * cw1[h * 3 + 1];   // coeff of w
        a2 += w2 * cw1[h * 3 + 2];   // coeff of vj
        cc += w2 * cb1[h];
    }
    cc += cb2[0];
    const float srate = RATE_CONST / (float)batch_num[0];

    __syncthreads();

    // ---- stage 3: z-tile = W_tile(16xK) @ v^T(Kx16) via WMMA f32 16x16x4 ----
    // A frag: lane L(0..15) -> W[oBase+L, k+0..1]; lane L+16 -> W[oBase+L, k+2..3]
    // B frag: lane N(0..15) -> v[N, k+0..1];       lane N+16 -> v[N, k+2..3]
    // Batch has only 8 rows; B columns 8..15 replicate rows 0..7. That garbage
    // stays confined to D columns 8..15 (never read), so no masking is needed.
    const int laneM = lane & 15;
    const int kHalf = (lane >> 4) * 2;
    const int kLen  = K >> 2;              // per-wave K chunk (scalar)
    const int k0    = wave * kLen;         // scalar

    const float* __restrict__ aPtr = W + (size_t)(oBase + laneM) * K + kHalf;
    const float* __restrict__ bPtr = lds_v + (laneM & 7) * LDSK + kHalf;

    v8f acc0 = {};
    v8f acc1 = {};
#pragma unroll 4
    for (int k = k0; k < k0 + kLen; k += 8) {
        v2f aA = *(const v2f*)(aPtr + k);
        v2f bA = *(const v2f*)(bPtr + k);
        acc0 = __builtin_amdgcn_wmma_f32_16x16x4_f32(
            false, aA, false, bA, (short)0, acc0, false, false);
        v2f aB = *(const v2f*)(aPtr + k + 4);
        v2f bB = *(const v2f*)(bPtr + k + 4);
        acc1 = __builtin_amdgcn_wmma_f32_16x16x4_f32(
            false, aB, false, bB, (short)0, acc1, false, false);
    }
    lds_part[wave][lane] = acc0 + acc1;
    __syncthreads();

    // ---- stage 4: reduce 4 partial tiles + fused Hebbian epilogue ----
    // D layout: lane n(0..15): VGPR r = D[M=r, N=n]; lane n+16: D[M=r+8, N=n]
    if (wave == 0) {
        v8f tot = lds_part[0][lane] + lds_part[1][lane]
                + lds_part[2][lane] + lds_part[3][lane];
        const int m  = lane & 15;            // batch row
        const int hi = (lane >> 4) * 8;      // +8 o-row offset for lanes 16..31
        if (m < 8) {
            const float S = lds_S[m];
            const float Q = lds_Q[m];
            const float vjScale = 1.f + srate * a2 * S;
            const float constT  = srate * (a0 * Q + cc * S);
#pragma unroll
            for (int r = 0; r < 8; ++r) {
                const int   o  = oBase + r + hi;
                const float z  = tot[r];
                const float vj = fmaxf(z + bias[o], 0.f);
                out[m * O + o] = vj * vjScale + srate * a1 * z + constT;
            }
        }
    }
}

extern "C" void kernel_launch(void* const* d_in, const int* in_sizes, int n_in,
                              void* d_out, int out_size, void* d_ws, size_t ws_size,
                              hipStream_t stream) {
    const float* x   = (const float*)d_in[0];
    const float* W1  = (const float*)d_in[1];
    const float* b1  = (const float*)d_in[2];
    const float* W2  = (const float*)d_in[3];
    const float* b2  = (const float*)d_in[4];
    const float* W3  = (const float*)d_in[5];
    const float* b3  = (const float*)d_in[6];
    const float* cw1 = (const float*)d_in[7];
    const float* cb1 = (const float*)d_in[8];
    const float* cw2 = (const float*)d_in[9];
    const float* cb2 = (const float*)d_in[10];
    const int*   bn  = (const int*)d_in[11];

    float* out = (float*)d_out;
    float* ws  = (float*)d_ws;
    float* h1o = ws;                 // (8, 1024)
    float* h2o = ws + 8 * 1024;      // (8, 1024)

    dim3 blk(128);
    hebb_layer_kernel<<<dim3(1024 / 16), blk, 0, stream>>>(
        x,   W1, b1, cw1, cb1, cw2, cb2, bn, h1o, 1024, 1024);
    hebb_layer_kernel<<<dim3(1024 / 16), blk, 0, stream>>>(
        h1o, W2, b2, cw1, cb1, cw2, cb2, bn, h2o, 1024, 1024);
    hebb_layer_kernel<<<dim3(512 / 16),  blk, 0, stream>>>(
        h2o, W3, b3, cw1, cb1, cw2, cb2, bn, out, 1024, 512);
}